// HeterogeneousGraphAttention_75007308858101
// MI455X (gfx1250) — compile-verified
//
#include <hip/hip_runtime.h>
#include <cfloat>
#include <cmath>

#define N_NODES 100000
#define N_USER  50000
#define D_IN    100
#define F_OUT   16
#define N_HEAD  3
#define N_TYPES 3
#define N_EDGES 1600000
#define D2      128
#define NHF     (N_HEAD * F_OUT)        /* 48 */
#define ETOT    (N_EDGES + N_NODES)     /* edges + self loops = 1,700,000 */

typedef float v2f __attribute__((ext_vector_type(2)));
typedef float v8f __attribute__((ext_vector_type(8)));

// ---------------------------------------------------------------------------
// Wave-level fp32 WMMA tile: C(16x16) = A(16xK) * B(Kx16), K % 4 == 0.
// A 16x4 frag layout (ISA 7.12.2): lanes 0-15 rows, VGPR0={K+0 | K+2}, VGPR1={K+1 | K+3}.
// C/D layout: VGPR v -> M = v + 8*(lane>=16), N = lane&15.
// ---------------------------------------------------------------------------
__device__ __forceinline__ v8f wmma_f32_tile(const float* __restrict__ A, int lda,
                                             const float* __restrict__ B, int ldb, int K) {
  const int lane = threadIdx.x & 31;
  const int half = lane >> 4;     // 0: lanes 0-15, 1: lanes 16-31
  const int rc   = lane & 15;     // row for A, col for B
  v8f c = {};
  for (int k = 0; k < K; k += 4) {
    v2f a, b;
    const float* ap = A + (size_t)rc * lda + k + 2 * half;
    a.x = ap[0];
    a.y = ap[1];
    const float* bp = B + (size_t)(k + 2 * half) * ldb + rc;
    b.x = bp[0];
    b.y = bp[ldb];
    c = __builtin_amdgcn_wmma_f32_16x16x4_f32(false, a, false, b, (short)0, c,
                                              false, false);
  }
  return c;
}

// ---------------------------------------------------------------------------
// hp[t][n][48] = h(100000x100) @ W[t](100x48) + b[t]  -- one 16x16 tile / wave
// ---------------------------------------------------------------------------
__global__ void gemm_hp_kernel(const float* __restrict__ h, const float* __restrict__ W,
                               const float* __restrict__ bias, float* __restrict__ hp) {
  const int MT = N_NODES / 16;               // 6250
  const int NT = (N_TYPES * NHF) / 16;       // 9
  int wave = blockIdx.x * (blockDim.x >> 5) + (threadIdx.x >> 5);
  if (wave >= MT * NT) return;
  int mt = wave / NT, nt = wave % NT;
  int m0 = mt * 16, n0 = nt * 16;
  int t = n0 / NHF, j0 = n0 % NHF;           // 48 % 16 == 0: tile never spans types
  const float* Bp = W + (size_t)t * D_IN * NHF + j0;
  v8f c = wmma_f32_tile(h + (size_t)m0 * D_IN, D_IN, Bp, NHF, D_IN);
  int lane = threadIdx.x & 31, col = lane & 15, half = lane >> 4;
  float bb = bias[t * NHF + j0 + col];
  float* Cp = hp + (size_t)t * N_NODES * NHF + (size_t)(m0 + 8 * half) * NHF + j0 + col;
#pragma unroll
  for (int v = 0; v < 8; ++v) Cp[(size_t)v * NHF] = c[v] + bb;
}

// ---------------------------------------------------------------------------
// q1[n][128] = h[:50000](50000x100) @ w1(100x128)
// ---------------------------------------------------------------------------
__global__ void gemm_q1_kernel(const float* __restrict__ h, const float* __restrict__ w1,
                               float* __restrict__ q1) {
  const int MT = N_USER / 16;                // 3125
  const int NT = D2 / 16;                    // 8
  int wave = blockIdx.x * (blockDim.x >> 5) + (threadIdx.x >> 5);
  if (wave >= MT * NT) return;
  int mt = wave / NT, nt = wave % NT;
  int m0 = mt * 16, n0 = nt * 16;
  v8f c = wmma_f32_tile(h + (size_t)m0 * D_IN, D_IN, w1 + n0, D2, D_IN);
  int lane = threadIdx.x & 31, col = lane & 15, half = lane >> 4;
  float* Cp = q1 + (size_t)(m0 + 8 * half) * D2 + n0 + col;
#pragma unroll
  for (int v = 0; v < 8; ++v) Cp[(size_t)v * D2] = c[v];
}

// ---------------------------------------------------------------------------
// Zero accumulators, set per-type running max to -FLT_MAX.
// ---------------------------------------------------------------------------
__global__ void init_kernel(float* __restrict__ gat, float* __restrict__ denom,
                            float* __restrict__ emax) {
  size_t i = (size_t)blockIdx.x * blockDim.x + threadIdx.x;
  const size_t ngat = (size_t)N_TYPES * N_USER * F_OUT;
  const size_t nden = (size_t)N_TYPES * N_USER * N_HEAD;
  if (i < ngat) gat[i] = 0.0f;
  if (i < nden) denom[i] = 0.0f;
  if (i < N_TYPES) emax[i] = -FLT_MAX;
}

// ---------------------------------------------------------------------------
// Pass 1 (per type): e = leaky_relu(hs . (a_src+a_trg)) per head; store e for
// edges whose target is a user node; block-reduce global max (all edges).
// ---------------------------------------------------------------------------
__global__ void edge_e_kernel(const float* __restrict__ hp_t,
                              const int* __restrict__ src_idx,
                              const int* __restrict__ trg_idx,
                              const float* __restrict__ a_src_t,
                              const float* __restrict__ a_trg_t,
                              float* __restrict__ ebuf, float* __restrict__ emax_t) {
  __shared__ float s_a[NHF];
  __shared__ float s_red[256];
  int tid = threadIdx.x;
  if (tid < NHF) s_a[tid] = a_src_t[tid] + a_trg_t[tid];
  __syncthreads();

  size_t i = (size_t)blockIdx.x * blockDim.x + tid;
  float lmax = -FLT_MAX;
  if (i < (size_t)ETOT) {
    int src, trg;
    if (i < (size_t)N_EDGES) { src = src_idx[i]; trg = trg_idx[i]; }
    else                     { src = (int)(i - N_EDGES); trg = src; }
    const float* hs = hp_t + (size_t)src * NHF;
    bool keep = (trg < N_USER);
#pragma unroll
    for (int hh = 0; hh < N_HEAD; ++hh) {
      float s = 0.0f;
#pragma unroll
      for (int f = 0; f < F_OUT; ++f) s += hs[hh * F_OUT + f] * s_a[hh * F_OUT + f];
      s = (s > 0.0f) ? s : 0.2f * s;                 // leaky_relu(0.2)
      if (keep) ebuf[(size_t)hh * ETOT + i] = s;
      lmax = fmaxf(lmax, s);
    }
  }
  s_red[tid] = lmax;
  __syncthreads();
  for (int off = 128; off > 0; off >>= 1) {
    if (tid < off) s_red[tid] = fmaxf(s_red[tid], s_red[tid + off]);
    __syncthreads();
  }
  if (tid == 0) {                                     // float atomic-max trick
    float v = s_red[0];
    if (v >= 0.0f) atomicMax((int*)emax_t, __float_as_int(v));
    else           atomicMin((unsigned int*)emax_t, __float_as_uint(v));
  }
}

// ---------------------------------------------------------------------------
// Pass 2 (per type): exp_e = exp(e - emax); denom[trg][h] += exp_e  (user trg only)
// ---------------------------------------------------------------------------
__global__ void edge_exp_kernel(const int* __restrict__ trg_idx,
                                const float* __restrict__ emax_t,
                                float* __restrict__ ebuf, float* __restrict__ denom_t) {
  size_t i = (size_t)blockIdx.x * blockDim.x + threadIdx.x;
  if (i >= (size_t)ETOT) return;
  int trg = (i < (size_t)N_EDGES) ? trg_idx[i] : (int)(i - N_EDGES);
  if (trg >= N_USER) return;                          // result never read
  float mx = *emax_t;
#pragma unroll
  for (int hh = 0; hh < N_HEAD; ++hh) {
    float ex = expf(ebuf[(size_t)hh * ETOT + i] - mx);
    ebuf[(size_t)hh * ETOT + i] = ex;
    atomicAdd(denom_t + (size_t)trg * N_HEAD + hh, ex);
  }
}

// ---------------------------------------------------------------------------
// Pass 3 (per type): alpha = exp_e / (denom+1e-10); gat[trg][f] += mean_h(alpha*hs)
// Heads folded in registers -> 16 atomics/edge, 1/3 head-mean folded in.
// ---------------------------------------------------------------------------
__global__ void edge_scatter_kernel(const float* __restrict__ hp_t,
                                    const int* __restrict__ src_idx,
                                    const int* __restrict__ trg_idx,
                                    const float* __restrict__ ebuf,
                                    const float* __restrict__ denom_t,
                                    float* __restrict__ gat_t) {
  size_t i = (size_t)blockIdx.x * blockDim.x + threadIdx.x;
  if (i >= (size_t)ETOT) return;
  int src, trg;
  if (i < (size_t)N_EDGES) { src = src_idx[i]; trg = trg_idx[i]; }
  else                     { src = (int)(i - N_EDGES); trg = src; }
  if (trg >= N_USER) return;
  const float* hs = hp_t + (size_t)src * NHF;
  float alpha[N_HEAD];
#pragma unroll
  for (int hh = 0; hh < N_HEAD; ++hh)
    alpha[hh] = ebuf[(size_t)hh * ETOT + i] /
                (denom_t[(size_t)trg * N_HEAD + hh] + 1e-10f);
  const float inv = 1.0f / (float)N_HEAD;
#pragma unroll
  for (int f = 0; f < F_OUT; ++f) {
    float c = 0.0f;
#pragma unroll
    for (int hh = 0; hh < N_HEAD; ++hh) c += alpha[hh] * hs[hh * F_OUT + f];
    atomicAdd(gat_t + (size_t)trg * F_OUT + f, c * inv);
  }
}

// ---------------------------------------------------------------------------
// Final semantic-attention fusion: one block (128 threads = D2) per user node.
// q = tanh(q1[n] + ta[n,r] @ w2); score_r = q . m; beta = softmax; fuse.
// out[n] = [fusion(16) | ta[0](16) | ta[1](16) | ta[2](16)]
// ---------------------------------------------------------------------------
__global__ void fusion_kernel(const float* __restrict__ q1, const float* __restrict__ gat,
                              const float* __restrict__ w2, const float* __restrict__ m,
                              float* __restrict__ out) {
  int n = blockIdx.x;
  int d = threadIdx.x;                                // 0..127
  __shared__ float s_ta[N_TYPES * F_OUT];
  __shared__ float s_red[D2];
  __shared__ float s_beta[N_TYPES];

  if (d < N_TYPES * F_OUT) {
    int r = d / F_OUT, f = d % F_OUT;
    s_ta[d] = gat[(size_t)r * N_USER * F_OUT + (size_t)n * F_OUT + f];
  }
  __syncthreads();

  float q1v = q1[(size_t)n * D2 + d];
  float mv = m[d];
  float w2v[F_OUT];
#pragma unroll
  for (int f = 0; f < F_OUT; ++f) w2v[f] = w2[f * D2 + d];

  float scores[N_TYPES];
#pragma unroll
  for (int r = 0; r < N_TYPES; ++r) {
    float acc = q1v;
#pragma unroll
    for (int f = 0; f < F_OUT; ++f) acc += s_ta[r * F_OUT + f] * w2v[f];
    scores[r] = tanhf(acc) * mv;
  }
#pragma unroll
  for (int r = 0; r < N_TYPES; ++r) {                 // 128-wide sum reductions
    s_red[d] = scores[r];
    __syncthreads();
    for (int off = 64; off > 0; off >>= 1) {
      if (d < off) s_red[d] += s_red[d + off];
      __syncthreads();
    }
    if (d == 0) s_beta[r] = s_red[0];
    __syncthreads();
  }
  if (d == 0) {
    float smax = fmaxf(s_beta[0], fmaxf(s_beta[1], s_beta[2]));
    float e0 = expf(s_beta[0] - smax), e1 = expf(s_beta[1] - smax),
          e2 = expf(s_beta[2] - smax);
    float inv = 1.0f / (e0 + e1 + e2);
    s_beta[0] = e0 * inv; s_beta[1] = e1 * inv; s_beta[2] = e2 * inv;
  }
  __syncthreads();

  float* outn = out + (size_t)n * 4 * F_OUT;
  if (d < F_OUT) {
    outn[d] = s_beta[0] * s_ta[d] + s_beta[1] * s_ta[F_OUT + d] +
              s_beta[2] * s_ta[2 * F_OUT + d];
  }
  if (d < N_TYPES * F_OUT) outn[F_OUT + d] = s_ta[d];
}

// ---------------------------------------------------------------------------
extern "C" void kernel_launch(void* const* d_in, const int* in_sizes, int n_in,
                              void* d_out, int out_size, void* d_ws, size_t ws_size,
                              hipStream_t stream) {
  const float* h     = (const float*)d_in[0];
  const float* W     = (const float*)d_in[1];
  const float* b     = (const float*)d_in[2];
  const float* a_src = (const float*)d_in[3];
  const float* a_trg = (const float*)d_in[4];
  const float* w1    = (const float*)d_in[5];
  const float* w2    = (const float*)d_in[6];
  const float* m     = (const float*)d_in[7];
  const int*   ei    = (const int*)d_in[8];           // (3,2,N_EDGES)

  float* ws = (float*)d_ws;
  size_t off = 0;
  float* hp   = ws + off; off += (size_t)N_TYPES * N_NODES * NHF;   // 14.4M f
  float* q1   = ws + off; off += (size_t)N_USER * D2;               //  6.4M f
  float* gat  = ws + off; off += (size_t)N_TYPES * N_USER * F_OUT;  //  2.4M f
  float* den  = ws + off; off += (size_t)N_TYPES * N_USER * N_HEAD; //  0.45M f
  float* ebuf = ws + off; off += (size_t)N_HEAD * ETOT;             //  5.1M f (reused per type)
  float* emax = ws + off; off += 4;

  {   // zero accumulators + seed max
    size_t nmax = (size_t)N_TYPES * N_USER * F_OUT;
    init_kernel<<<(int)((nmax + 255) / 256), 256, 0, stream>>>(gat, den, emax);
  }
  {   // h @ W + b  (WMMA f32 16x16x4)
    int waves = (N_NODES / 16) * ((N_TYPES * NHF) / 16);
    gemm_hp_kernel<<<(waves + 3) / 4, 128, 0, stream>>>(h, W, b, hp);
  }
  {   // h[:N_USER] @ w1  (WMMA f32 16x16x4)
    int waves = (N_USER / 16) * (D2 / 16);
    gemm_q1_kernel<<<(waves + 3) / 4, 128, 0, stream>>>(h, w1, q1);
  }

  int eblocks = (ETOT + 255) / 256;
  for (int t = 0; t < N_TYPES; ++t) {
    const float* hp_t = hp + (size_t)t * N_NODES * NHF;
    const int* src_t = ei + (size_t)t * 2 * N_EDGES;
    const int* trg_t = src_t + N_EDGES;
    float* den_t = den + (size_t)t * N_USER * N_HEAD;
    edge_e_kernel<<<eblocks, 256, 0, stream>>>(hp_t, src_t, trg_t,
                                               a_src + (size_t)t * NHF,
                                               a_trg + (size_t)t * NHF,
                                               ebuf, emax + t);
    edge_exp_kernel<<<eblocks, 256, 0, stream>>>(trg_t, emax + t, ebuf, den_t);
    edge_scatter_kernel<<<eblocks, 256, 0, stream>>>(hp_t, src_t, trg_t, ebuf, den_t,
                                                     gat + (size_t)t * N_USER * F_OUT);
  }

  fusion_kernel<<<N_USER, 128, 0, stream>>>(q1, gat, w2, m, (float*)d_out);
}